// ToMeLayer_52269751992931
// MI455X (gfx1250) — compile-verified
//
#include <hip/hip_runtime.h>
#include <stdint.h>

#define B_   4
#define T_   4096
#define D_   2048
#define R_   512
#define EPS_ 1e-12f

typedef __attribute__((ext_vector_type(2))) float v2f;
typedef __attribute__((ext_vector_type(8))) float v8f;

// ---------------------------------------------------------------------------
// Kernel 1: inverse L2 norms of batch-0 rows.  inv[i] = 1/max(||x0[i]||, EPS)
// ---------------------------------------------------------------------------
__global__ void tome_norms(const float* __restrict__ x, float* __restrict__ inv) {
  __shared__ float red[256];
  const int row = blockIdx.x;
  const float* p = x + (size_t)row * D_;
  float s = 0.f;
  for (int d = threadIdx.x; d < D_; d += 256) { float v = p[d]; s += v * v; }
  red[threadIdx.x] = s;
  __syncthreads();
  for (int off = 128; off > 0; off >>= 1) {
    if (threadIdx.x < (unsigned)off) red[threadIdx.x] += red[threadIdx.x + off];
    __syncthreads();
  }
  if (threadIdx.x == 0) inv[row] = 1.0f / fmaxf(sqrtf(red[0]), EPS_);
}

// ---------------------------------------------------------------------------
// Kernel 2: adjacent cosine similarities via V_WMMA_F32_16X16X4_F32.
// One wave per block.  Tile t computes adj[16t .. 16t+15] as the diagonal of
// C = A * B, A = rows [base..base+15], B^T = rows [base+1..base+16],
// accumulated over K = D in steps of 4 (f32 WMMA shape 16x16x4).
// A layout (ISA 7.12.2): lanes 0-15 -> K={0,1} in v[0],v[1]; lanes 16-31 -> K={2,3}.
// B layout mirrors with N = lane&15.  Both reduce to the same per-lane load:
// float2 at column (k + 2*(lane>>4)) of the lane's row.
// ---------------------------------------------------------------------------
__global__ void tome_adj_wmma(const float* __restrict__ x,
                              const float* __restrict__ inv,
                              float* __restrict__ adj) {
  const int lane = threadIdx.x;        // wave32, one wave per block
  const int base = blockIdx.x * 16;
  const int m  = lane & 15;
  const int hi = lane >> 4;            // 0: K pair {0,1}, 1: K pair {2,3}
  int rowA = base + m;     if (rowA > T_ - 1) rowA = T_ - 1;
  int rowB = base + 1 + m; if (rowB > T_ - 1) rowB = T_ - 1;
  const float* pa = x + (size_t)rowA * D_ + 2 * hi;
  const float* pb = x + (size_t)rowB * D_ + 2 * hi;

  v8f acc = {};
#pragma unroll 8
  for (int k = 0; k < D_; k += 4) {
    v2f a = *(const v2f*)(pa + k);
    v2f b = *(const v2f*)(pb + k);
    // (neg_a, A, neg_b, B, c_mod, C, reuse_a, reuse_b)
    acc = __builtin_amdgcn_wmma_f32_16x16x4_f32(false, a, false, b,
                                                (short)0, acc, false, false);
  }

  // Extract diagonal C[L][L]: lane = (L<8 ? L : L+16), VGPR = L&7.
  float dv = 0.f;
  int di = -1;
#pragma unroll
  for (int v = 0; v < 8; ++v) {
    if (lane == v)      { dv = acc[v]; di = base + v; }
    if (lane == v + 24) { dv = acc[v]; di = base + 8 + v; }
  }
  if (di >= 0 && di < T_ - 1)
    adj[di] = dv * inv[di] * inv[di + 1];
}

// ---------------------------------------------------------------------------
// Kernel 3: selection.  Single 1024-thread block, all state in LDS.
// (a) Iterated dominant-edge rounds == sequential greedy maximal matching
//     under priority (w desc, idx asc):  dom-left: w[i] >  w[i-1]
//                                        dom-right: w[i] >= w[i+1]
// (b) Top-R of the matching by packed 64-bit key, bitonic sort descending.
// Writes sel flags (int 0/1, length T_) to global scratch.
// ---------------------------------------------------------------------------
__global__ void tome_select(const float* __restrict__ adj, int* __restrict__ selg) {
  __shared__ float w[T_];
  __shared__ unsigned long long key[T_];
  __shared__ unsigned char alive[T_];
  __shared__ unsigned char selr[T_];
  __shared__ unsigned char matched[T_];
  __shared__ int anyAlive;
  const int tid = threadIdx.x;

  for (int i = tid; i < T_; i += 1024) {
    const bool valid = (i < T_ - 1);
    w[i] = valid ? adj[i] : -3.0e38f;
    alive[i] = valid ? 1 : 0;
    matched[i] = 0;
    selr[i] = 0;
  }
  __syncthreads();

  for (int round = 0; round < T_; ++round) {
    // phase 1: dominant edges (consistent snapshot of alive)
    for (int i = tid; i < T_ - 1; i += 1024) {
      bool s = false;
      if (alive[i]) {
        const bool dl = (i == 0)      || !alive[i - 1] || (w[i] >  w[i - 1]);
        const bool dr = (i == T_ - 2) || !alive[i + 1] || (w[i] >= w[i + 1]);
        s = dl && dr;
      }
      selr[i] = s ? 1 : 0;
    }
    __syncthreads();
    // phase 2: accept + kill neighbors (selected edges are never adjacent)
    for (int i = tid; i < T_ - 1; i += 1024) {
      if (selr[i]) {
        matched[i] = 1;
        alive[i] = 0;
        if (i > 0)      alive[i - 1] = 0;
        if (i < T_ - 2) alive[i + 1] = 0;
      }
    }
    if (tid == 0) anyAlive = 0;
    __syncthreads();
    bool loc = false;
    for (int i = tid; i < T_ - 1; i += 1024) loc |= (alive[i] != 0);
    if (loc) anyAlive = 1;
    __syncthreads();
    if (!anyAlive) break;
  }

  // keys: (order-preserving float map << 32) | (0xFFFFFFFF - idx)
  for (int i = tid; i < T_; i += 1024) {
    unsigned long long kk = 0ull;
    if (i < T_ - 1 && matched[i]) {
      unsigned int u = __float_as_uint(w[i]);
      unsigned int o = (u & 0x80000000u) ? ~u : (u | 0x80000000u);
      kk = ((unsigned long long)o << 32) |
           (unsigned long long)(0xFFFFFFFFu - (unsigned)i);
    }
    key[i] = kk;
  }
  __syncthreads();

  // bitonic sort, descending
  for (unsigned k = 2; k <= T_; k <<= 1) {
    for (unsigned j = k >> 1; j > 0; j >>= 1) {
      for (unsigned i = tid; i < T_; i += 1024) {
        const unsigned ixj = i ^ j;
        if (ixj > i) {
          const bool desc = ((i & k) == 0);
          unsigned long long a = key[i], b = key[ixj];
          if (desc ? (a < b) : (a > b)) { key[i] = b; key[ixj] = a; }
        }
      }
      __syncthreads();
    }
  }

  // reuse alive[] (all zero now) as final sel flags
  for (int i = tid; i < T_; i += 1024) alive[i] = 0;
  __syncthreads();
  if (tid < R_) {
    const unsigned idx = 0xFFFFFFFFu - (unsigned)(key[tid] & 0xFFFFFFFFull);
    alive[idx] = 1;
  }
  __syncthreads();
  for (int i = tid; i < T_; i += 1024) selg[i] = alive[i];
}

// ---------------------------------------------------------------------------
// Kernel 4: gather/unmerge.  One block per (b,t) row; float4 streaming.
//   j = (t>0 && sel[t-1]) ? t-1 : t
//   out[b,t] = sel[j] ? 0.5*(x[b,j]+x[b,j+1]) : x[b,j]
// ---------------------------------------------------------------------------
__global__ void tome_gather(const float* __restrict__ x,
                            const int* __restrict__ selg,
                            float* __restrict__ out) {
  const int bt = blockIdx.x;          // 0 .. B*T-1
  const int t  = bt & (T_ - 1);
  const int j  = (t > 0 && selg[t - 1]) ? (t - 1) : t;
  const bool mrg = (selg[j] != 0);    // selg[T_-1] == 0 always => j+1 in-bounds

  const size_t rowO = (size_t)bt * D_;
  const size_t rowJ = ((size_t)(bt - t) + (size_t)j) * D_;
  const float4* p0 = (const float4*)(x + rowJ);
  const float4* p1 = (const float4*)(x + rowJ + D_);
  float4* o = (float4*)(out + rowO);

  const int n4 = D_ / 4;
  for (int d = threadIdx.x; d < n4; d += 256) {
    float4 a = p0[d];
    if (mrg) {
      const float4 c = p1[d];
      a.x = 0.5f * (a.x + c.x);
      a.y = 0.5f * (a.y + c.y);
      a.z = 0.5f * (a.z + c.z);
      a.w = 0.5f * (a.w + c.w);
    }
    o[d] = a;
  }
}

// ---------------------------------------------------------------------------
extern "C" void kernel_launch(void* const* d_in, const int* in_sizes, int n_in,
                              void* d_out, int out_size, void* d_ws, size_t ws_size,
                              hipStream_t stream) {
  (void)in_sizes; (void)n_in; (void)out_size; (void)ws_size;
  const float* x  = (const float*)d_in[0];
  float* out      = (float*)d_out;
  float* ws       = (float*)d_ws;
  float* inv      = ws;                   // T_ floats
  float* adj      = ws + T_;              // T_ floats (adj[T_-1] unused)
  int*   selg     = (int*)(ws + 2 * T_);  // T_ ints

  tome_norms   <<<T_,        256, 0, stream>>>(x, inv);
  tome_adj_wmma<<<T_ / 16,    32, 0, stream>>>(x, inv, adj);
  tome_select  <<<1,        1024, 0, stream>>>(adj, selg);
  tome_gather  <<<B_ * T_,   256, 0, stream>>>(x, selg, out);
}